// ProtoMiner_59433757442365
// MI455X (gfx1250) — compile-verified
//
#include <hip/hip_runtime.h>
#include <math.h>

#define N_  131072
#define D_  256
#define K_  32
#define B_  1024
#define M_  128

typedef float v2f __attribute__((ext_vector_type(2)));
typedef float v8f __attribute__((ext_vector_type(8)));

static __device__ __forceinline__ v8f wmma4(v2f a, v2f b, v8f c) {
  // D = A(16x4 f32) * B(4x16 f32) + C(16x16 f32)
  return __builtin_amdgcn_wmma_f32_16x16x4_f32(false, a, false, b, (short)0, c, false, false);
}

// ---------------- zero init for BN stats ----------------
__global__ void k_zero_stats(float* stats) {
  stats[threadIdx.x] = 0.f; // 512 threads: sum[256], sumsq[256]
}

// ---------------- GEMM1: h1 = z @ W1 + b1  (wave = 16 rows x 256 cols) ----
__global__ __launch_bounds__(256) void k_gemm1(const float* __restrict__ z,
                                               const float* __restrict__ W1,
                                               const float* __restrict__ b1,
                                               float* __restrict__ h1) {
  int wv = threadIdx.x >> 5, lane = threadIdx.x & 31;
  int half = lane >> 4, lm = lane & 15;
  int mt = blockIdx.x * 8 + wv;                 // 0..8191 row tiles
  const float* zrow = z + (size_t)(mt * 16 + lm) * D_;
  v8f zero = {0.f,0.f,0.f,0.f,0.f,0.f,0.f,0.f};
  v8f c[16];
#pragma unroll
  for (int t = 0; t < 16; ++t) c[t] = zero;
  for (int k0 = 0; k0 < D_; k0 += 4) {
    int ka = k0 + half * 2;
    v2f a = *(const v2f*)(zrow + ka);
    const float* wa = W1 + (size_t)ka * D_ + lm;
#pragma unroll
    for (int t = 0; t < 16; ++t) {
      v2f b; b.x = wa[t * 16]; b.y = wa[D_ + t * 16];
      c[t] = wmma4(a, b, c[t]);
    }
  }
  int rbase = mt * 16 + half * 8;
#pragma unroll
  for (int t = 0; t < 16; ++t) {
    int col = t * 16 + lm;
    float bias = b1[col];
#pragma unroll
    for (int v = 0; v < 8; ++v)
      h1[(size_t)(rbase + v) * D_ + col] = c[t][v] + bias;
  }
}

// ---------------- per-feature sum / sumsq over N rows ----------------
__global__ __launch_bounds__(256) void k_stats(const float* __restrict__ h1,
                                               float* __restrict__ stats) {
  int d = threadIdx.x;
  size_t r0 = (size_t)blockIdx.x * 256;
  float s1 = 0.f, s2 = 0.f;
  for (int r = 0; r < 256; ++r) {
    float v = h1[(r0 + r) * D_ + d];
    s1 += v; s2 += v * v;
  }
  atomicAdd(&stats[d], s1);
  atomicAdd(&stats[256 + d], s2);
}

// ---------------- BN scale/shift ----------------
__global__ void k_bnparams(const float* __restrict__ stats,
                           const float* __restrict__ gamma,
                           const float* __restrict__ beta,
                           float* __restrict__ sc) {
  int d = threadIdx.x; // 256
  float mu  = stats[d] * (1.f / (float)N_);
  float var = stats[256 + d] * (1.f / (float)N_) - mu * mu;
  float inv = rsqrtf(var + 1e-5f);
  float s = gamma[d] * inv;
  sc[d] = s;
  sc[256 + d] = beta[d] - mu * s;
}

// ---------------- GEMM2: logits = lrelu(BN(h1)) @ W2 + b2 ----------------
__global__ __launch_bounds__(256) void k_gemm2(const float* __restrict__ h1,
                                               const float* __restrict__ sc,
                                               const float* __restrict__ W2,
                                               const float* __restrict__ b2,
                                               float* __restrict__ logits) {
  int wv = threadIdx.x >> 5, lane = threadIdx.x & 31;
  int half = lane >> 4, lm = lane & 15;
  int mt = blockIdx.x * 8 + wv;
  const float* scale = sc;
  const float* shift = sc + 256;
  const float* hrow = h1 + (size_t)(mt * 16 + lm) * D_;
  v8f zero = {0.f,0.f,0.f,0.f,0.f,0.f,0.f,0.f};
  v8f c[2] = {zero, zero};
  for (int k0 = 0; k0 < D_; k0 += 4) {
    int ka = k0 + half * 2;
    float x0 = hrow[ka]     * scale[ka]     + shift[ka];
    float x1 = hrow[ka + 1] * scale[ka + 1] + shift[ka + 1];
    x0 = x0 > 0.f ? x0 : 0.01f * x0;
    x1 = x1 > 0.f ? x1 : 0.01f * x1;
    v2f a; a.x = x0; a.y = x1;
#pragma unroll
    for (int t = 0; t < 2; ++t) {
      int col = t * 16 + lm;
      v2f b; b.x = W2[ka * K_ + col]; b.y = W2[(ka + 1) * K_ + col];
      c[t] = wmma4(a, b, c[t]);
    }
  }
  int rbase = mt * 16 + half * 8;
#pragma unroll
  for (int t = 0; t < 2; ++t) {
    int col = t * 16 + lm;
    float bias = b2[col];
#pragma unroll
    for (int v = 0; v < 8; ++v)
      logits[(size_t)(rbase + v) * K_ + col] = c[t][v] + bias;
  }
}

// ---------------- softmax(T=0.1) + std(ddof=1) filter; wave per row --------
__global__ __launch_bounds__(256) void k_softmax(float* __restrict__ logits,
                                                 float* __restrict__ s_out) {
  int wv = threadIdx.x >> 5, lane = threadIdx.x & 31;
  size_t r = (size_t)blockIdx.x * 8 + wv;
  float lg = logits[r * K_ + lane] * 10.0f;  // /T1
  float mx = lg;
#pragma unroll
  for (int off = 16; off >= 1; off >>= 1) mx = fmaxf(mx, __shfl_xor(mx, off, 32));
  float e = __expf(lg - mx);
  float tot = e;
#pragma unroll
  for (int off = 16; off >= 1; off >>= 1) tot += __shfl_xor(tot, off, 32);
  float sv = e / tot;
  s_out[r * K_ + lane] = sv;
  float ssum = sv;
#pragma unroll
  for (int off = 16; off >= 1; off >>= 1) ssum += __shfl_xor(ssum, off, 32);
  float mean = ssum * (1.f / 32.f);
  float dv = sv - mean;
  float var = dv * dv;
#pragma unroll
  for (int off = 16; off >= 1; off >>= 1) var += __shfl_xor(var, off, 32);
  var *= (1.f / 31.f);
  float sd = sqrtf(var);
  float filt = sv > (sd + (1.f / 32.f)) ? sv : 0.f;
  logits[r * K_ + lane] = filt;              // sf written in place
}

// ---------------- per-graph: p_hat, L1-normalized pooling, mask -----------
__global__ __launch_bounds__(256) void k_pool(const float* __restrict__ sf,
                                              const float* __restrict__ z,
                                              float* __restrict__ hdense,
                                              float* __restrict__ hflat,
                                              float* __restrict__ phat,
                                              float* __restrict__ submask) {
  __shared__ float sfl[M_ * K_];
  __shared__ float Sv[K_], invl1[K_], absrow[K_], maskv[K_];
  int tid = threadIdx.x;
  int b = blockIdx.x;
  for (int i = tid; i < M_ * K_; i += 256) sfl[i] = sf[(size_t)b * M_ * K_ + i];
  __syncthreads();
  if (tid < K_) {
    float s = 0.f;
    for (int m = 0; m < M_; ++m) s += sfl[m * K_ + tid];
    Sv[tid] = s;
    invl1[tid] = 1.f / fmaxf(s, 1e-12f);   // sf >= 0 so L1 sum == S
    absrow[tid] = 0.f;
  }
  __syncthreads();
  for (int i = tid; i < K_ * K_; i += 256) {
    int r = i >> 5, cc = i & 31;
    float v = (r == cc) ? (Sv[r] / (Sv[r] + 1e-12f)) : 0.f;
    phat[(size_t)b * K_ * K_ + i] = v;
  }
  int wv = tid >> 5, lane = tid & 31, half = lane >> 4, lm = lane & 15;
  int rt = wv & 1, cg = wv >> 1;             // 2 row tiles x 4 col groups
  int arow = rt * 16 + lm;
  float ai = invl1[arow];
  const float* zb = z + (size_t)b * M_ * D_;
  v8f zero = {0.f,0.f,0.f,0.f,0.f,0.f,0.f,0.f};
  v8f c[4] = {zero, zero, zero, zero};
  for (int k0 = 0; k0 < M_; k0 += 4) {
    int ka = k0 + half * 2;
    v2f a; a.x = sfl[ka * K_ + arow] * ai; a.y = sfl[(ka + 1) * K_ + arow] * ai;
#pragma unroll
    for (int t = 0; t < 4; ++t) {
      int col = cg * 64 + t * 16 + lm;
      v2f bb; bb.x = zb[ka * D_ + col]; bb.y = zb[(ka + 1) * D_ + col];
      c[t] = wmma4(a, bb, c[t]);
    }
  }
  int rbase = rt * 16 + half * 8;
#pragma unroll
  for (int t = 0; t < 4; ++t) {
    int col = cg * 64 + t * 16 + lm;
#pragma unroll
    for (int v = 0; v < 8; ++v) {
      float val = c[t][v];
      hdense[(size_t)b * K_ * D_ + (size_t)(rbase + v) * D_ + col] = val;
      atomicAdd(&absrow[rbase + v], fabsf(val));
    }
  }
  __syncthreads();
  if (tid < K_) {
    float mk = absrow[tid] > 0.f ? 1.f : 0.f;
    maskv[tid] = mk;
    submask[(size_t)b * K_ + tid] = mk;
  }
  __syncthreads();
#pragma unroll
  for (int t = 0; t < 4; ++t) {
    int col = cg * 64 + t * 16 + lm;
#pragma unroll
    for (int v = 0; v < 8; ++v)
      hflat[(size_t)b * K_ * D_ + (size_t)(rbase + v) * D_ + col] =
          c[t][v] * maskv[rbase + v];
  }
}

// ---------------- sizes[k] = sum_j mask.flat[k*B + j] (reshape semantics) --
__global__ void k_sizes(const float* __restrict__ submask, float* __restrict__ sizes) {
  int k = threadIdx.x; // 32
  float s = 0.f;
  for (int j = 0; j < B_; ++j) s += submask[(size_t)k * B_ + j];
  sizes[k] = s;
}

// ---------------- U_new = g*U + (1-g)*centroid ----------------
__global__ __launch_bounds__(256) void k_unew(const float* __restrict__ hdense,
                                              const float* __restrict__ sizes,
                                              const float* __restrict__ U,
                                              float* __restrict__ Unew) {
  int k = blockIdx.x, d = threadIdx.x;
  float s = 0.f;
  for (int b = 0; b < B_; ++b) s += hdense[(size_t)b * K_ * D_ + (size_t)k * D_ + d];
  float cen = s / (sizes[k] + 1e-10f);
  Unew[k * D_ + d] = 0.99f * U[k * D_ + d] + 0.01f * cen;
}

// ---------------- row-normalized U ----------------
__global__ __launch_bounds__(256) void k_unorm(const float* __restrict__ U,
                                               float* __restrict__ Un) {
  int wv = threadIdx.x >> 5, lane = threadIdx.x & 31;
  for (int row = wv; row < K_; row += 8) {
    float ss = 0.f;
#pragma unroll
    for (int j = 0; j < 8; ++j) { float v = U[row * D_ + lane + j * 32]; ss += v * v; }
#pragma unroll
    for (int off = 16; off >= 1; off >>= 1) ss += __shfl_xor(ss, off, 32);
    float inv = 1.f / fmaxf(sqrtf(ss), 1e-12f);
#pragma unroll
    for (int j = 0; j < 8; ++j)
      Un[row * D_ + lane + j * 32] = U[row * D_ + lane + j * 32] * inv;
  }
}

// ---------------- inverse norms of h_flat rows ----------------
__global__ __launch_bounds__(256) void k_invn(const float* __restrict__ hflat,
                                              float* __restrict__ invn) {
  int wv = threadIdx.x >> 5, lane = threadIdx.x & 31;
  size_t row = (size_t)blockIdx.x * 8 + wv;
  float ss = 0.f;
#pragma unroll
  for (int j = 0; j < 8; ++j) { float v = hflat[row * D_ + lane + j * 32]; ss += v * v; }
#pragma unroll
  for (int off = 16; off >= 1; off >>= 1) ss += __shfl_xor(ss, off, 32);
  invn[row] = 1.f / fmaxf(sqrtf(ss), 1e-12f);
}

// ---------------- p = hn @ Un^T ----------------
__global__ __launch_bounds__(256) void k_pgemm(const float* __restrict__ hflat,
                                               const float* __restrict__ invn,
                                               const float* __restrict__ Un,
                                               float* __restrict__ p) {
  int wv = threadIdx.x >> 5, lane = threadIdx.x & 31;
  int half = lane >> 4, lm = lane & 15;
  int mt = blockIdx.x * 8 + wv;            // 0..2047
  int row = mt * 16 + lm;
  float rin = invn[row];
  const float* hr = hflat + (size_t)row * D_;
  v8f zero = {0.f,0.f,0.f,0.f,0.f,0.f,0.f,0.f};
  v8f c[2] = {zero, zero};
  for (int k0 = 0; k0 < D_; k0 += 4) {
    int ka = k0 + half * 2;
    v2f a; a.x = hr[ka] * rin; a.y = hr[ka + 1] * rin;
#pragma unroll
    for (int t = 0; t < 2; ++t) {
      int col = t * 16 + lm;
      v2f bb; bb.x = Un[col * D_ + ka]; bb.y = Un[col * D_ + ka + 1];
      c[t] = wmma4(a, bb, c[t]);
    }
  }
  int rbase = mt * 16 + half * 8;
#pragma unroll
  for (int t = 0; t < 2; ++t)
#pragma unroll
    for (int v = 0; v < 8; ++v)
      p[(size_t)(rbase + v) * K_ + t * 16 + lm] = c[t][v];
}

extern "C" void kernel_launch(void* const* d_in, const int* in_sizes, int n_in,
                              void* d_out, int out_size, void* d_ws, size_t ws_size,
                              hipStream_t stream) {
  (void)in_sizes; (void)n_in; (void)out_size; (void)ws_size;
  const float* z   = (const float*)d_in[0];
  const float* g   = (const float*)d_in[1];
  // d_in[2] = batch (arange//M) -> structure is a pure reshape, unused
  const float* W1  = (const float*)d_in[3];
  const float* b1  = (const float*)d_in[4];
  const float* bng = (const float*)d_in[5];
  const float* bnb = (const float*)d_in[6];
  const float* W2  = (const float*)d_in[7];
  const float* b2  = (const float*)d_in[8];
  const float* U   = (const float*)d_in[9];

  float* out = (float*)d_out;
  const size_t off_s      = 0;
  const size_t off_hflat  = off_s      + (size_t)N_ * K_;
  const size_t off_hdense = off_hflat  + (size_t)B_ * K_ * D_;
  const size_t off_p      = off_hdense + (size_t)B_ * K_ * D_;
  const size_t off_phat   = off_p      + (size_t)B_ * K_ * K_;
  const size_t off_zd     = off_phat   + (size_t)B_ * K_ * K_;
  const size_t off_g      = off_zd     + (size_t)N_ * D_;
  const size_t off_mask   = off_g      + (size_t)B_ * D_;
  const size_t off_unew   = off_mask   + (size_t)B_ * K_;

  float* s_out   = out + off_s;
  float* hflat   = out + off_hflat;
  float* hdense  = out + off_hdense;
  float* p_out   = out + off_p;
  float* phat    = out + off_phat;
  float* zdense  = out + off_zd;
  float* gout    = out + off_g;
  float* submask = out + off_mask;
  float* unew    = out + off_unew;

  float* ws     = (float*)d_ws;
  float* logits = ws;                          // N*K floats, reused as sf
  float* stats  = ws + (size_t)N_ * K_;        // sum[256],sumsq[256],scale[256],shift[256]
  float* sizes  = stats + 1024;                // 32
  float* Un     = sizes + 32;                  // K*D
  float* invn   = Un + (size_t)K_ * D_;        // B*K

  // Reuse the z_dense output region as h1 scratch; real copy lands after GEMM2.
  float* h1 = zdense;

  hipLaunchKernelGGL(k_zero_stats, dim3(1), dim3(512), 0, stream, stats);
  hipLaunchKernelGGL(k_gemm1, dim3(1024), dim3(256), 0, stream, z, W1, b1, h1);
  hipLaunchKernelGGL(k_stats, dim3(512), dim3(256), 0, stream, h1, stats);
  hipLaunchKernelGGL(k_bnparams, dim3(1), dim3(256), 0, stream, stats, bng, bnb, stats + 512);
  hipLaunchKernelGGL(k_gemm2, dim3(1024), dim3(256), 0, stream, h1, stats + 512, W2, b2, logits);
  // h1 no longer needed: overwrite z_dense with the real reshape copy
  hipMemcpyAsync(zdense, z, (size_t)N_ * D_ * sizeof(float), hipMemcpyDeviceToDevice, stream);
  hipMemcpyAsync(gout, g, (size_t)B_ * D_ * sizeof(float), hipMemcpyDeviceToDevice, stream);
  hipLaunchKernelGGL(k_softmax, dim3(N_ / 8), dim3(256), 0, stream, logits, s_out);
  hipLaunchKernelGGL(k_pool, dim3(B_), dim3(256), 0, stream, logits, z, hdense, hflat, phat, submask);
  hipLaunchKernelGGL(k_sizes, dim3(1), dim3(32), 0, stream, submask, sizes);
  hipLaunchKernelGGL(k_unew, dim3(K_), dim3(256), 0, stream, hdense, sizes, U, unew);
  hipLaunchKernelGGL(k_unorm, dim3(1), dim3(256), 0, stream, U, Un);
  hipLaunchKernelGGL(k_invn, dim3(B_ * K_ / 8), dim3(256), 0, stream, hflat, invn);
  hipLaunchKernelGGL(k_pgemm, dim3(B_ * K_ / 128), dim3(256), 0, stream, hflat, invn, Un, p_out);
}